// KANClassification_4157528343109
// MI455X (gfx1250) — compile-verified
//
#include <hip/hip_runtime.h>
#include <hip/hip_bf16.h>

typedef float v2f __attribute__((ext_vector_type(2)));
typedef float v8f __attribute__((ext_vector_type(8)));

// ---------------- workspace layout (float offsets, 64B-aligned) ----------------
#define WS_C1W     0         // [2][27][9]   = 486
#define WS_C2W     496       // [2][18][9]   = 324
#define WS_W1      832       // pair-interleaved [576][64][2] = 73728
#define WS_W2      74560     // pair-interleaved [288][16][2] = 9216
#define WS_POOLED  83776     // [256][2][10][10] = 51200
#define WS_CONV2   134976    // [256][2][8][8]   = 32768
#define WS_STATS   167744    // [2][2] mean,inv  = 4
#define WS_F1      167760    // [256][1152]  = 294912
#define WS_F2      462672    // [256][576]   = 147456

// ---------------- SiLU + cubic B-spline features (grid 5, order 3) ----------------
// knots t[j] = (j-3)*0.4 - 1, j=0..11 ; 8 degree-3 bases via Cox–de Boor
__device__ __forceinline__ void kan_features(float x, float* f9) {
    const float h = 0.4f;
    float t[12];
#pragma unroll
    for (int j = 0; j < 12; ++j) t[j] = (float)(j - 3) * h - 1.0f;
    float b[11];
#pragma unroll
    for (int m = 0; m < 11; ++m) b[m] = (x >= t[m] && x < t[m + 1]) ? 1.0f : 0.0f;
#pragma unroll
    for (int deg = 1; deg <= 3; ++deg) {
        float inv = 1.0f / ((float)deg * h);
#pragma unroll
        for (int m = 0; m < 11 - deg; ++m)
            b[m] = ((x - t[m]) * b[m] + (t[m + deg + 1] - x) * b[m + 1]) * inv;
    }
    f9[0] = x / (1.0f + __expf(-x));   // silu
#pragma unroll
    for (int g = 0; g < 8; ++g) f9[1 + g] = b[g];
}

// ---------------- 0. weight preprocessing ----------------
// W1/W2 are stored pair-interleaved over K so a WMMA B-fragment (B[k][n], B[k+1][n])
// is one contiguous 8-byte load:  Wp[(k>>1)*2N + n*2 + (k&1)]
__global__ __launch_bounds__(256) void prep_weights(
    const float* __restrict__ c1_base, const float* __restrict__ c1_spline, const float* __restrict__ c1_scaler,
    const float* __restrict__ c2_base, const float* __restrict__ c2_spline, const float* __restrict__ c2_scaler,
    const float* __restrict__ k1_base, const float* __restrict__ k1_spline, const float* __restrict__ k1_scaler,
    const float* __restrict__ k2_base, const float* __restrict__ k2_spline, const float* __restrict__ k2_scaler,
    float* __restrict__ ws)
{
    int t = blockIdx.x * blockDim.x + threadIdx.x;
    if (t < 54) {                       // c1 reduced weights: [2][27][9]
        int o = t / 27, i = t % 27;
        float s0 = 0.f, sg[8] = {0.f,0.f,0.f,0.f,0.f,0.f,0.f,0.f};
        for (int j = 0; j < 27; ++j) {
            int idx = (o * 27 + j) * 27 + i;
            s0 += c1_base[idx];
            float sc = c1_scaler[idx];
#pragma unroll
            for (int g = 0; g < 8; ++g) sg[g] += c1_spline[idx * 8 + g] * sc;
        }
        float* w = ws + WS_C1W + (o * 27 + i) * 9;
        w[0] = s0;
#pragma unroll
        for (int g = 0; g < 8; ++g) w[1 + g] = sg[g];
    } else if ((t -= 54) < 36) {        // c2 reduced weights: [2][18][9]
        int o = t / 18, i = t % 18;
        float s0 = 0.f, sg[8] = {0.f,0.f,0.f,0.f,0.f,0.f,0.f,0.f};
        for (int j = 0; j < 18; ++j) {
            int idx = (o * 18 + j) * 18 + i;
            s0 += c2_base[idx];
            float sc = c2_scaler[idx];
#pragma unroll
            for (int g = 0; g < 8; ++g) sg[g] += c2_spline[idx * 8 + g] * sc;
        }
        float* w = ws + WS_C2W + (o * 18 + i) * 9;
        w[0] = s0;
#pragma unroll
        for (int g = 0; g < 8; ++g) w[1 + g] = sg[g];
    } else if ((t -= 36) < 8192) {      // W1: K=1152 (k = i*9+f, i<128), N=64
        int o = t & 63, i = t >> 6;
        int idx = o * 128 + i;
        float* w = ws + WS_W1;
        float sc = k1_scaler[idx];
        float vals[9];
        vals[0] = k1_base[idx];
#pragma unroll
        for (int g = 0; g < 8; ++g) vals[1 + g] = k1_spline[idx * 8 + g] * sc;
#pragma unroll
        for (int f = 0; f < 9; ++f) {
            int k = i * 9 + f;
            w[(k >> 1) * 128 + o * 2 + (k & 1)] = vals[f];
        }
    } else if ((t -= 8192) < 1024) {    // W2: K=576 (k = i*9+f, i<64), N=16, cols 10..15 zero
        int n = t & 15, i = t >> 4;     // i < 64
        float* w = ws + WS_W2;
        float vals[9];
        if (n < 10) {
            int idx = n * 64 + i;
            float sc = k2_scaler[idx];
            vals[0] = k2_base[idx];
#pragma unroll
            for (int g = 0; g < 8; ++g) vals[1 + g] = k2_spline[idx * 8 + g] * sc;
        } else {
#pragma unroll
            for (int f = 0; f < 9; ++f) vals[f] = 0.0f;
        }
#pragma unroll
        for (int f = 0; f < 9; ++f) {
            int k = i * 9 + f;
            w[(k >> 1) * 32 + n * 2 + (k & 1)] = vals[f];
        }
    }
}

// ---------------- 1. KAN conv1 (27->2, sum over Dout) fused with 3x3 avg-pool ----------------
__global__ __launch_bounds__(256) void conv1_pool(const float* __restrict__ x, const float* __restrict__ ws_c,
                                                  float* __restrict__ pooled)
{
    __shared__ float feat[32 * 32 * 9];   // one channel of per-pixel features
    __shared__ float w[2 * 27 * 9];
    int b = blockIdx.x, tid = threadIdx.x;
    for (int i = tid; i < 2 * 27 * 9; i += 256) w[i] = ws_c[WS_C1W + i];

    bool active = tid < 200;
    int o = tid / 100, p = tid % 100, ph = p / 10, pw = p % 10;
    float acc = 0.0f;
    for (int c = 0; c < 3; ++c) {
        __syncthreads();
        for (int q = tid; q < 1024; q += 256) {
            float v = x[((b * 3 + c) * 32) * 32 + q];
            kan_features(v, &feat[q * 9]);
        }
        __syncthreads();
        if (active) {
            for (int dh = 0; dh < 3; ++dh)
                for (int dw = 0; dw < 3; ++dw) {
                    int h0 = 3 * ph + dh, w0 = 3 * pw + dw;   // conv output position
#pragma unroll
                    for (int kh = 0; kh < 3; ++kh)
#pragma unroll
                        for (int kw = 0; kw < 3; ++kw) {
                            const float* f  = &feat[((h0 + kh) * 32 + (w0 + kw)) * 9];
                            const float* wv = &w[(o * 27 + (c * 9 + kh * 3 + kw)) * 9];
#pragma unroll
                            for (int j = 0; j < 9; ++j) acc += f[j] * wv[j];
                        }
                }
        }
    }
    if (active) pooled[((b * 2 + o) * 10 + ph) * 10 + pw] = acc * (1.0f / 9.0f);
}

// ---------------- 2. KAN conv2 (18->2, sum over Dout) ----------------
__global__ __launch_bounds__(256) void conv2(const float* __restrict__ ws_in, float* __restrict__ conv2out)
{
    __shared__ float feat[2 * 10 * 10 * 9];
    __shared__ float w[2 * 18 * 9];
    int b = blockIdx.x, tid = threadIdx.x;
    for (int i = tid; i < 2 * 18 * 9; i += 256) w[i] = ws_in[WS_C2W + i];
    if (tid < 200) {
        float v = ws_in[WS_POOLED + b * 200 + tid];
        kan_features(v, &feat[tid * 9]);
    }
    __syncthreads();
    if (tid < 128) {
        int o = tid >> 6, q = tid & 63, hh = q >> 3, wwp = q & 7;
        float acc = 0.0f;
        for (int c = 0; c < 2; ++c)
#pragma unroll
            for (int kh = 0; kh < 3; ++kh)
#pragma unroll
                for (int kw = 0; kw < 3; ++kw) {
                    const float* f  = &feat[((c * 10 + hh + kh) * 10 + (wwp + kw)) * 9];
                    const float* wv = &w[(o * 18 + c * 9 + kh * 3 + kw) * 9];
#pragma unroll
                    for (int j = 0; j < 9; ++j) acc += f[j] * wv[j];
                }
        conv2out[b * 128 + o * 64 + q] = acc;
    }
}

// ---------------- 3. batch-norm statistics (deterministic block reduce) ----------------
__global__ __launch_bounds__(256) void bn_stats(const float* __restrict__ conv2out, float* __restrict__ stats)
{
    __shared__ float rs[256], rs2[256];
    int o = blockIdx.x, tid = threadIdx.x;
    float s = 0.f, s2 = 0.f;
    for (int idx = tid; idx < 16384; idx += 256) {
        int b = idx >> 6, pos = idx & 63;
        float v = conv2out[b * 128 + o * 64 + pos];
        s += v; s2 += v * v;
    }
    rs[tid] = s; rs2[tid] = s2;
    __syncthreads();
    for (int off = 128; off > 0; off >>= 1) {
        if (tid < off) { rs[tid] += rs[tid + off]; rs2[tid] += rs2[tid + off]; }
        __syncthreads();
    }
    if (tid == 0) {
        float mean = rs[0] * (1.0f / 16384.0f);
        float var  = rs2[0] * (1.0f / 16384.0f) - mean * mean;
        stats[o * 2 + 0] = mean;
        stats[o * 2 + 1] = rsqrtf(var + 1e-5f);
    }
}

// ---------------- 4. normalize + emit F1 features [256][1152] ----------------
__global__ __launch_bounds__(256) void bn_feat(const float* __restrict__ conv2out, const float* __restrict__ stats,
                                               const float* __restrict__ gamma, const float* __restrict__ beta,
                                               float* __restrict__ F1)
{
    int idx = blockIdx.x * 256 + threadIdx.x;   // 32768 = 256*128
    int b = idx >> 7, f = idx & 127, c = f >> 6;
    float v = (conv2out[idx] - stats[c * 2 + 0]) * stats[c * 2 + 1] * gamma[c] + beta[c];
    float f9[9];
    kan_features(v, f9);
    float* dst = F1 + b * 1152 + f * 9;
#pragma unroll
    for (int j = 0; j < 9; ++j) dst[j] = f9[j];
}

// ---------------- 5. WMMA fp32 GEMM [256,1152]x[1152,64] -> features F2 ----------------
// B is pair-interleaved: Bp[(k>>1)*128 + n*2 + (k&1)] -> one b64 load per fragment.
__global__ __launch_bounds__(32) void gemm1_wmma(const float* __restrict__ A, const float* __restrict__ Bp,
                                                 float* __restrict__ F2)
{
    const int K = 1152;
    int m0 = blockIdx.x * 16, n0 = blockIdx.y * 16;
    int lane = threadIdx.x, half = lane >> 4, r = lane & 15;
    const float* arow  = A + (m0 + r) * K + 2 * half;     // (A[k+2h], A[k+2h+1]) contiguous
    const float* bbase = Bp + half * 128 + (n0 + r) * 2;  // pair (k0/2 + half) at +(k0>>1)*128
    v8f acc = {};
#pragma unroll 8
    for (int k = 0; k < K; k += 4) {
        v2f a  = *(const v2f*)(arow + k);
        v2f bf = *(const v2f*)(bbase + (k >> 1) * 128);
        acc = __builtin_amdgcn_wmma_f32_16x16x4_f32(false, a, false, bf, (short)0, acc, false, false);
    }
    // epilogue: D[rr] is (M = rr + 8*half, N = r); build next layer's KAN features
#pragma unroll
    for (int rr = 0; rr < 8; ++rr) {
        int row = m0 + rr + 8 * half;
        int col = n0 + r;
        float f9[9];
        kan_features(acc[rr], f9);
        float* dst = F2 + row * 576 + col * 9;
#pragma unroll
        for (int j = 0; j < 9; ++j) dst[j] = f9[j];
    }
}

// ---------------- 6. WMMA fp32 GEMM [256,576]x[576,16] -> out [256,10] ----------------
__global__ __launch_bounds__(32) void gemm2_wmma(const float* __restrict__ A, const float* __restrict__ Bp,
                                                 float* __restrict__ out)
{
    const int K = 576;
    int m0 = blockIdx.x * 16;
    int lane = threadIdx.x, half = lane >> 4, r = lane & 15;
    const float* arow  = A + (m0 + r) * K + 2 * half;
    const float* bbase = Bp + half * 32 + r * 2;
    v8f acc = {};
#pragma unroll 8
    for (int k = 0; k < K; k += 4) {
        v2f a  = *(const v2f*)(arow + k);
        v2f bf = *(const v2f*)(bbase + (k >> 1) * 32);
        acc = __builtin_amdgcn_wmma_f32_16x16x4_f32(false, a, false, bf, (short)0, acc, false, false);
    }
#pragma unroll
    for (int rr = 0; rr < 8; ++rr) {
        int row = m0 + rr + 8 * half;
        if (r < 10) out[row * 10 + r] = acc[rr];
    }
}

extern "C" void kernel_launch(void* const* d_in, const int* in_sizes, int n_in,
                              void* d_out, int out_size, void* d_ws, size_t ws_size,
                              hipStream_t stream) {
    const float* x         = (const float*)d_in[0];
    const float* c1_base   = (const float*)d_in[1];
    const float* c1_spline = (const float*)d_in[2];
    const float* c1_scaler = (const float*)d_in[3];
    const float* c2_base   = (const float*)d_in[4];
    const float* c2_spline = (const float*)d_in[5];
    const float* c2_scaler = (const float*)d_in[6];
    const float* bn_gamma  = (const float*)d_in[7];
    const float* bn_beta   = (const float*)d_in[8];
    const float* k1_base   = (const float*)d_in[9];
    const float* k1_spline = (const float*)d_in[10];
    const float* k1_scaler = (const float*)d_in[11];
    const float* k2_base   = (const float*)d_in[12];
    const float* k2_spline = (const float*)d_in[13];
    const float* k2_scaler = (const float*)d_in[14];
    float* ws  = (float*)d_ws;
    float* out = (float*)d_out;

    prep_weights<<<37, 256, 0, stream>>>(c1_base, c1_spline, c1_scaler,
                                         c2_base, c2_spline, c2_scaler,
                                         k1_base, k1_spline, k1_scaler,
                                         k2_base, k2_spline, k2_scaler, ws);
    conv1_pool<<<256, 256, 0, stream>>>(x, ws, ws + WS_POOLED);
    conv2<<<256, 256, 0, stream>>>(ws, ws + WS_CONV2);
    bn_stats<<<2, 256, 0, stream>>>(ws + WS_CONV2, ws + WS_STATS);
    bn_feat<<<128, 256, 0, stream>>>(ws + WS_CONV2, ws + WS_STATS, bn_gamma, bn_beta, ws + WS_F1);
    gemm1_wmma<<<dim3(16, 4), 32, 0, stream>>>(ws + WS_F1, ws + WS_W1, ws + WS_F2);
    gemm2_wmma<<<dim3(16, 1), 32, 0, stream>>>(ws + WS_F2, ws + WS_W2, out);
}